// GAT_44882408243317
// MI455X (gfx1250) — compile-verified
//
#include <hip/hip_runtime.h>
#include <hip/hip_bf16.h>

#define NN 50000
#define EE 800000
#define EE2 (NN + EE)          // 850000 edges incl. self loops
#define FIN 2
#define FE  7
#define NH  4
#define NC  32
#define HC  128
#define RH  500

typedef __attribute__((ext_vector_type(2))) float v2f;
typedef __attribute__((ext_vector_type(8))) float v8f;

static inline int divup(int a, int b) { return (a + b - 1) / b; }

__device__ __forceinline__ v8f wmma4(v2f a, v2f b, v8f c) {
    return __builtin_amdgcn_wmma_f32_16x16x4_f32(false, a, false, b, (short)0, c,
                                                 false, false);
}

// ---------------- small utility kernels ----------------

__global__ void fill_f32(float* p, int n, float v) {
    int i = blockIdx.x * blockDim.x + threadIdx.x;
    if (i < n) p[i] = v;
}

// p[r*cols + c] = bias[c]
__global__ void fill_rowbias(float* p, int rows, int cols, const float* __restrict__ bias) {
    int i = blockIdx.x * blockDim.x + threadIdx.x;
    if (i < rows * cols) p[i] = bias[i % cols];
}

// ---------------- self-loop mean edge-attr ----------------

__global__ void selfloop_sum(const int* __restrict__ ei, const float* __restrict__ ea,
                             float* deg, float* mea) {
    int e = blockIdx.x * blockDim.x + threadIdx.x;
    if (e >= EE) return;
    int d = ei[EE + e];
    atomicAdd(&deg[d], 1.0f);
    #pragma unroll
    for (int k = 0; k < FE; ++k)
        atomicAdd(&mea[(size_t)d * FE + k], ea[(size_t)e * FE + k]);
}

__global__ void selfloop_div(float* mea, const float* __restrict__ deg) {
    int i = blockIdx.x * blockDim.x + threadIdx.x;
    if (i >= NN) return;
    float dd = fmaxf(deg[i], 1.0f);
    #pragma unroll
    for (int k = 0; k < FE; ++k) mea[(size_t)i * FE + k] /= dd;
}

// ---------------- WMMA f32 GEMM: C = post(pre(A) @ W + bias) ----------------
// A: MxK row-major, W: KxNd row-major, bias: Nd.
// One wave computes a 16x32 C tile (two 16x16 N-tiles sharing the A fragment).
// OOB rows/cols are index-CLAMPED (not zeroed): an A lane only feeds its own C
// row and a B lane only feeds its own C column, both masked at the store.

__global__ void wmma_gemm(const float* __restrict__ A, const float* __restrict__ W,
                          const float* __restrict__ bias, float* __restrict__ Cout,
                          int M, int K, int Nd,
                          int preact, float preslope, int postact, float postslope) {
    int wave = blockIdx.x * (blockDim.x >> 5) + (threadIdx.x >> 5);
    int tnc = (Nd + 31) >> 5;                 // 32-wide N tiles
    int tmc = (M + 15) >> 4;
    if (wave >= tmc * tnc) return;            // wave-uniform exit: EXEC all-1 below
    int tm = wave / tnc, tn = wave % tnc;
    int lane = threadIdx.x & 31;
    int half = lane >> 4, l16 = lane & 15;
    int kbase = 2 * half;                     // lanes 0-15 hold K=k..k+1, 16-31 K=k+2..k+3
    int row = tm * 16 + l16;
    int rowc = row < M ? row : M - 1;
    int n0 = tn * 32 + l16;
    int n1 = n0 + 16;
    int n0c = n0 < Nd ? n0 : Nd - 1;
    int n1c = n1 < Nd ? n1 : Nd - 1;
    const float* Ap = A + (size_t)rowc * K + kbase;
    v8f c0 = {}, c1 = {};

    if ((K & 3) == 0) {                       // fast path: K multiple of 4 (128, 500)
        for (int k = 0; k < K; k += 4) {
            v2f a = *(const v2f*)(Ap + k);    // 8B-aligned contiguous pair
            if (preact) {
                a[0] = a[0] > 0.f ? a[0] : a[0] * preslope;
                a[1] = a[1] > 0.f ? a[1] : a[1] * preslope;
            }
            const float* wk = W + (size_t)(k + kbase) * Nd;
            v2f b0, b1;
            b0[0] = wk[n0c];       b0[1] = wk[Nd + n0c];
            b1[0] = wk[n1c];       b1[1] = wk[Nd + n1c];
            c0 = wmma4(a, b0, c0);            // two independent accumulation chains
            c1 = wmma4(a, b1, c1);
        }
    } else {                                  // slow path: layer 0 only (K=2)
        for (int k = 0; k < K; k += 4) {
            v2f a, b0, b1;
            #pragma unroll
            for (int j = 0; j < 2; ++j) {
                int kk = k + kbase + j;
                int kc = kk < K ? kk : K - 1;
                float av = A[(size_t)rowc * K + kc];
                if (preact) av = av > 0.f ? av : av * preslope;
                a[j] = av;
                const float* wk = W + (size_t)kc * Nd;
                float m0 = kk < K ? 1.f : 0.f;  // zero B so padded K contributes 0
                b0[j] = wk[n0c] * m0;
                b1[j] = wk[n1c] * m0;
            }
            c0 = wmma4(a, b0, c0);
            c1 = wmma4(a, b1, c1);
        }
    }

    #pragma unroll
    for (int i = 0; i < 8; ++i) {
        int m = tm * 16 + i + 8 * half;       // C layout: VGPR i -> M=i / M=8+i
        if (m < M) {
            if (n0 < Nd) {
                float v = c0[i] + bias[n0];
                if (postact && v < 0.f) v *= postslope;
                Cout[(size_t)m * Nd + n0] = v;
            }
            if (n1 < Nd) {
                float v = c1[i] + bias[n1];
                if (postact && v < 0.f) v *= postslope;
                Cout[(size_t)m * Nd + n1] = v;
            }
        }
    }
}

// Fused final reghead stage: out[m] += sum_n wend[n] * leaky01(A@W2 + b2)[m,n]
// (K = 500, multiple of 4 -> fast path only)
__global__ void wmma_gemm_end(const float* __restrict__ A, const float* __restrict__ W,
                              const float* __restrict__ bias, const float* __restrict__ wend,
                              float* __restrict__ out, int M, int K, int Nd) {
    int wave = blockIdx.x * (blockDim.x >> 5) + (threadIdx.x >> 5);
    int tnc = (Nd + 31) >> 5;
    int tmc = (M + 15) >> 4;
    if (wave >= tmc * tnc) return;
    int tm = wave / tnc, tn = wave % tnc;
    int lane = threadIdx.x & 31;
    int half = lane >> 4, l16 = lane & 15;
    int kbase = 2 * half;
    int row = tm * 16 + l16;
    int rowc = row < M ? row : M - 1;
    int n0 = tn * 32 + l16;
    int n1 = n0 + 16;
    int n0c = n0 < Nd ? n0 : Nd - 1;
    int n1c = n1 < Nd ? n1 : Nd - 1;
    const float* Ap = A + (size_t)rowc * K + kbase;
    v8f c0 = {}, c1 = {};
    for (int k = 0; k < K; k += 4) {
        v2f a = *(const v2f*)(Ap + k);
        const float* wk = W + (size_t)(k + kbase) * Nd;
        v2f b0, b1;
        b0[0] = wk[n0c];       b0[1] = wk[Nd + n0c];
        b1[0] = wk[n1c];       b1[1] = wk[Nd + n1c];
        c0 = wmma4(a, b0, c0);
        c1 = wmma4(a, b1, c1);
    }
    #pragma unroll
    for (int i = 0; i < 8; ++i) {
        int m = tm * 16 + i + 8 * half;
        if (m < M) {
            float acc = 0.f;
            if (n0 < Nd) {
                float v = c0[i] + bias[n0];
                if (v < 0.f) v *= 0.01f;
                acc += v * wend[n0];
            }
            if (n1 < Nd) {
                float v = c1[i] + bias[n1];
                if (v < 0.f) v *= 0.01f;
                acc += v * wend[n1];
            }
            atomicAdd(&out[m], acc);
        }
    }
}

// ---------------- edge kernels ----------------

__device__ __forceinline__ void edge_sd(const int* __restrict__ ei, int e, int& s, int& d) {
    if (e < EE) { s = ei[e]; d = ei[EE + e]; }
    else        { s = d = e - EE; }
}

__device__ __forceinline__ void atomicMaxF(float* addr, float v) {
    if (v >= 0.f) atomicMax((int*)addr, __float_as_int(v));
    else          atomicMin((unsigned int*)addr, __float_as_uint(v));
}

// logits[e][h] = sum_c leaky0.2(xl[s]+xr[d]+ea@We)[h,c] * att[h,c]; track per-dst max
__global__ void edge_logits(const int* __restrict__ ei, const float* __restrict__ ea,
                            const float* __restrict__ mea,
                            const float* __restrict__ xl, const float* __restrict__ xr,
                            const float* __restrict__ We, const float* __restrict__ att,
                            float* __restrict__ lg, float* __restrict__ mmax) {
    int idx = blockIdx.x * blockDim.x + threadIdx.x;
    if (idx >= EE2 * NH) return;
    int e = idx >> 2, hh = idx & 3;
    int s, d; edge_sd(ei, e, s, d);
    const float* ep = (e < EE) ? (ea + (size_t)e * FE) : (mea + (size_t)(e - EE) * FE);
    float eav[FE];
    #pragma unroll
    for (int k = 0; k < FE; ++k) eav[k] = ep[k];
    const float* xls = xl + (size_t)s * HC + hh * NC;
    const float* xrd = xr + (size_t)d * HC + hh * NC;
    const float* ah  = att + hh * NC;
    float acc = 0.f;
    #pragma unroll
    for (int cidx = 0; cidx < NC; ++cidx) {
        float eec = 0.f;
        #pragma unroll
        for (int k = 0; k < FE; ++k) eec += eav[k] * We[k * HC + hh * NC + cidx];
        float v = xls[cidx] + xrd[cidx] + eec;
        v = v > 0.f ? v : 0.2f * v;
        acc += v * ah[cidx];
    }
    lg[idx] = acc;
    atomicMaxF(&mmax[d * NH + hh], acc);
}

__global__ void edge_exp(const int* __restrict__ ei, float* __restrict__ lg,
                         const float* __restrict__ mmax, float* __restrict__ dn) {
    int idx = blockIdx.x * blockDim.x + threadIdx.x;
    if (idx >= EE2 * NH) return;
    int e = idx >> 2, hh = idx & 3;
    int s, d; edge_sd(ei, e, s, d);
    float mm = mmax[d * NH + hh];
    if (mm < -1e37f) mm = 0.f;   // mirrors reference's isfinite guard
    float ex = __expf(lg[idx] - mm);
    lg[idx] = ex;
    atomicAdd(&dn[d * NH + hh], ex);
}

__global__ void edge_aggr(const int* __restrict__ ei, const float* __restrict__ lg,
                          const float* __restrict__ dn, const float* __restrict__ xl,
                          float* __restrict__ hn) {
    int idx = blockIdx.x * blockDim.x + threadIdx.x;
    if (idx >= EE2 * NH) return;
    int e = idx >> 2, hh = idx & 3;
    int s, d; edge_sd(ei, e, s, d);
    float al = lg[idx] / fmaxf(dn[d * NH + hh], 1e-16f);
    const float* xls = xl + (size_t)s * HC + hh * NC;
    float* hd = hn + (size_t)d * HC + hh * NC;
    #pragma unroll
    for (int cidx = 0; cidx < NC; ++cidx)
        atomicAdd(&hd[cidx], al * xls[cidx]);
}

// ---------------- launcher ----------------

extern "C" void kernel_launch(void* const* d_in, const int* in_sizes, int n_in,
                              void* d_out, int out_size, void* d_ws, size_t ws_size,
                              hipStream_t stream) {
    (void)in_sizes; (void)n_in; (void)out_size; (void)ws_size;

    const float* x  = (const float*)d_in[0];
    const int*   ei = (const int*)d_in[1];
    const float* ea = (const float*)d_in[2];
    // layer l params at 3 + 7*l: Wl, bl, Wr, br, We, att, bias
    const float* W1   = (const float*)d_in[38];
    const float* b1   = (const float*)d_in[39];
    const float* W2   = (const float*)d_in[40];
    const float* b2   = (const float*)d_in[41];
    const float* Wend = (const float*)d_in[42];
    const float* bend = (const float*)d_in[43];
    float* out = (float*)d_out;

    // workspace carve-out (f32)
    float* ws  = (float*)d_ws;
    size_t off = 0;
    float* deg = ws + off; off += NN;
    float* mea = ws + off; off += (size_t)NN * FE;
    float* hA  = ws + off; off += (size_t)NN * HC;
    float* hB  = ws + off; off += (size_t)NN * HC;
    float* xl  = ws + off; off += (size_t)NN * HC;
    float* xr  = ws + off; off += (size_t)NN * HC;
    float* lg  = ws + off; off += (size_t)EE2 * NH;
    float* mm  = ws + off; off += (size_t)NN * NH;
    float* dn  = ws + off; off += (size_t)NN * NH;
    float* r1  = ws + off; off += (size_t)NN * RH;

    const int BT = 256;

    // self-loop mean edge attributes
    fill_f32<<<divup(NN, BT), BT, 0, stream>>>(deg, NN, 0.f);
    fill_f32<<<divup(NN * FE, BT), BT, 0, stream>>>(mea, NN * FE, 0.f);
    selfloop_sum<<<divup(EE, BT), BT, 0, stream>>>(ei, ea, deg, mea);
    selfloop_div<<<divup(NN, BT), BT, 0, stream>>>(mea, deg);

    const int eth = EE2 * NH;                    // 3.4M edge-head threads
    float* hcur = nullptr;
    for (int l = 0; l < 5; ++l) {
        const float* Acur = (l == 0) ? x : hcur;
        int K = (l == 0) ? FIN : HC;
        int pre = (l > 0) ? 1 : 0;               // inter-layer leaky(0.01) folded into A load
        const float* Wl   = (const float*)d_in[3 + 7 * l + 0];
        const float* bl   = (const float*)d_in[3 + 7 * l + 1];
        const float* Wr   = (const float*)d_in[3 + 7 * l + 2];
        const float* br   = (const float*)d_in[3 + 7 * l + 3];
        const float* We   = (const float*)d_in[3 + 7 * l + 4];
        const float* att  = (const float*)d_in[3 + 7 * l + 5];
        const float* bias = (const float*)d_in[3 + 7 * l + 6];
        float* hnext = (l & 1) ? hB : hA;        // l=0->hA, 1->hB, ... final in hA

        int tiles = divup(NN, 16) * divup(HC, 32);   // 16x32 tiles per wave
        int gblk  = divup(tiles, 4);                  // 4 waves / 128-thread block
        wmma_gemm<<<gblk, 128, 0, stream>>>(Acur, Wl, bl, xl, NN, K, HC, pre, 0.01f, 0, 0.f);
        wmma_gemm<<<gblk, 128, 0, stream>>>(Acur, Wr, br, xr, NN, K, HC, pre, 0.01f, 0, 0.f);

        fill_f32<<<divup(NN * NH, BT), BT, 0, stream>>>(mm, NN * NH, -3e38f);
        fill_f32<<<divup(NN * NH, BT), BT, 0, stream>>>(dn, NN * NH, 0.f);
        fill_rowbias<<<divup(NN * HC, BT), BT, 0, stream>>>(hnext, NN, HC, bias);

        edge_logits<<<divup(eth, BT), BT, 0, stream>>>(ei, ea, mea, xl, xr, We, att, lg, mm);
        edge_exp<<<divup(eth, BT), BT, 0, stream>>>(ei, lg, mm, dn);
        edge_aggr<<<divup(eth, BT), BT, 0, stream>>>(ei, lg, dn, xl, hnext);

        hcur = hnext;
    }

    // reghead: r1 = leaky01(h @ W1 + b1)
    {
        int tiles = divup(NN, 16) * divup(RH, 32);
        wmma_gemm<<<divup(tiles, 4), 128, 0, stream>>>(hcur, W1, b1, r1,
                                                       NN, HC, RH, 0, 0.f, 1, 0.01f);
    }
    // out = bend + leaky01(r1 @ W2 + b2) @ Wend   (fused, atomic accumulation)
    fill_rowbias<<<divup(NN, BT), BT, 0, stream>>>(out, NN, 1, bend);
    {
        int tiles = divup(NN, 16) * divup(RH, 32);
        wmma_gemm_end<<<divup(tiles, 4), 128, 0, stream>>>(r1, W2, b2, Wend, out,
                                                          NN, RH, RH);
    }
}